// SiamFCBNHead_14242111553572
// MI455X (gfx1250) — compile-verified
//
#include <hip/hip_runtime.h>
#include <hip/hip_bf16.h>
#include <stdint.h>

typedef float v2f __attribute__((ext_vector_type(2)));
typedef float v8f __attribute__((ext_vector_type(8)));

#define KC 64     // K-chunk staged in LDS
#define PM 72     // As row pitch (floats): %64==8 -> halves 16 banks apart; 16B-aligned rows
#define PQ 136    // Xs row pitch (floats): %64==8

// CDNA5 async DMA: global -> LDS, 16 bytes per lane, tracked by ASYNCcnt.
__device__ __forceinline__ void async_copy16(const float* gsrc, const float* ldst) {
    asm volatile("global_load_async_to_lds_b128 %0, %1, off"
                 :
                 : "v"((uint32_t)(uintptr_t)ldst), "v"(gsrc)
                 : "memory");
}
__device__ __forceinline__ void wait_async0() {
    asm volatile("s_wait_asynccnt 0x0" ::: "memory");
}

// ---------------------------------------------------------------------------
// Kernel 1: per-batch GEMM  P[b][m=64 offsets][q=1024 positions]
//           P = Z^T (64x512) * X (512x1024), f32 WMMA 16x16x4
// grid: 128 batches * 8 column-chunks of 128; block: 256 threads (8 waves)
// wave w: M-tile (w&3), 64-column half ((w>>2)*64) -> 4 N-tiles of 16
// LDS is k-major (contiguous rows -> async B128 DMA), double-buffered.
// ---------------------------------------------------------------------------
__global__ __launch_bounds__(256) void xcorr_gemm(
    const float* __restrict__ z,   // [128][512][64]  (c-major, offset fastest)
    const float* __restrict__ x,   // [128][512][1024]
    float* __restrict__ P)         // [128][64][1024]
{
    __shared__ float As[2 * KC * PM];   // As[buf][k][m] = z[b][k0+k][m]
    __shared__ float Xs[2 * KC * PQ];   // Xs[buf][k][qc] = x[b][k0+k][q0+qc]

    const int b    = blockIdx.x >> 3;
    const int q0   = (blockIdx.x & 7) * 128;
    const int t    = threadIdx.x;
    const int lane = t & 31;
    const int w    = t >> 5;
    const int mt   = w & 3;        // M tile 0..3
    const int nq   = w >> 2;       // column half 0..1
    const int l16  = lane & 15;
    const int hi   = lane >> 4;    // 0/1 -> K sub-pair select

    const float* zb = z + (size_t)b * (512 * 64);
    const float* xb = x + (size_t)b * (512 * 1024);

    v8f acc[4] = {};

    // issue one chunk's DMA: 4 + 8 async B128 per thread
    auto stage = [&](int k0, int buf) {
        float* AsB = &As[buf * KC * PM];
        float* XsB = &Xs[buf * KC * PQ];
        #pragma unroll
        for (int i = 0; i < 4; ++i) {
            int flat = t + i * 256;          // 0..1023
            int k    = flat >> 4;            // 0..63
            int m4   = (flat & 15) * 4;      // 0..60
            async_copy16(zb + (size_t)(k0 + k) * 64 + m4, &AsB[k * PM + m4]);
        }
        #pragma unroll
        for (int i = 0; i < 8; ++i) {
            int flat = t + i * 256;          // 0..2047
            int k    = flat >> 5;            // 0..63
            int q4   = (flat & 31) * 4;      // 0..124
            async_copy16(xb + (size_t)(k0 + k) * 1024 + q0 + q4, &XsB[k * PQ + q4]);
        }
    };

    stage(0, 0);

    for (int c = 0; c < 8; ++c) {
        wait_async0();          // this wave's DMA for chunk c has landed
        __syncthreads();        // everyone's has (they all waited too)

        if (c < 7) stage((c + 1) * KC, (c + 1) & 1);   // overlap DMA with compute

        const float* AsB = &As[(c & 1) * KC * PM];
        const float* XsB = &Xs[(c & 1) * KC * PQ];
        const int colA = mt * 16 + l16;

        #pragma unroll
        for (int ks = 0; ks < KC; ks += 4) {
            const int kb = ks + 2 * hi;
            v2f a;
            a.x = AsB[kb * PM + colA];        // fuses to ds_load_2addr_b32
            a.y = AsB[(kb + 1) * PM + colA];
            #pragma unroll
            for (int nt = 0; nt < 4; ++nt) {
                const int colB = nq * 64 + nt * 16 + l16;
                v2f bb;
                bb.x = XsB[kb * PQ + colB];
                bb.y = XsB[(kb + 1) * PQ + colB];
                acc[nt] = __builtin_amdgcn_wmma_f32_16x16x4_f32(
                    /*neg_a=*/false, a, /*neg_b=*/false, bb,
                    /*c_mod=*/(short)0, acc[nt],
                    /*reuse_a=*/false, /*reuse_b=*/false);
            }
        }
    }

    // --- store C fragments: VGPR r -> (M = r + 8*hi, N = l16) ---
    float* Pb = P + (size_t)b * (64 * 1024);
    #pragma unroll
    for (int nt = 0; nt < 4; ++nt) {
        const int col = q0 + nq * 64 + nt * 16 + l16;
        #pragma unroll
        for (int r = 0; r < 8; ++r) {
            const int row = mt * 16 + hi * 8 + r;
            Pb[(size_t)row * 1024 + col] = acc[nt][r];
        }
    }
}

// ---------------------------------------------------------------------------
// Kernel 2: shift-and-add gather + per-block BN partial sums
// ---------------------------------------------------------------------------
__global__ __launch_bounds__(256) void gather_bn(
    const float* __restrict__ P, float* __restrict__ out,
    float* __restrict__ blkSum, float* __restrict__ blkSumSq)
{
    const int b = blockIdx.x;
    const float* Pb = P + (size_t)b * 65536;
    float s1 = 0.f, s2 = 0.f;

    for (int p = threadIdx.x; p < 625; p += 256) {
        const int i = p / 25, j = p % 25;
        float y = 0.f;
        #pragma unroll
        for (int dy = 0; dy < 8; ++dy)
            #pragma unroll
            for (int dx = 0; dx < 8; ++dx)
                y += Pb[(dy * 8 + dx) * 1024 + (i + dy) * 32 + (j + dx)];
        out[b * 625 + p] = y;
        s1 += y;
        s2 += y * y;
    }

    __shared__ float r1[256], r2[256];
    r1[threadIdx.x] = s1;
    r2[threadIdx.x] = s2;
    __syncthreads();
    for (int s = 128; s > 0; s >>= 1) {
        if ((int)threadIdx.x < s) {
            r1[threadIdx.x] += r1[threadIdx.x + s];
            r2[threadIdx.x] += r2[threadIdx.x + s];
        }
        __syncthreads();
    }
    if (threadIdx.x == 0) { blkSum[b] = r1[0]; blkSumSq[b] = r2[0]; }
}

// ---------------------------------------------------------------------------
// Kernel 3: final reduction -> mean, rstd
// ---------------------------------------------------------------------------
__global__ __launch_bounds__(128) void bn_stats(
    const float* __restrict__ blkSum, const float* __restrict__ blkSumSq,
    float* __restrict__ stats)
{
    __shared__ float r1[128], r2[128];
    const int t = threadIdx.x;
    r1[t] = blkSum[t];
    r2[t] = blkSumSq[t];
    __syncthreads();
    for (int s = 64; s > 0; s >>= 1) {
        if (t < s) { r1[t] += r1[t + s]; r2[t] += r2[t + s]; }
        __syncthreads();
    }
    if (t == 0) {
        const float inv  = 1.0f / 80000.0f;
        const float mean = r1[0] * inv;
        const float var  = r2[0] * inv - mean * mean;
        stats[0] = mean;
        stats[1] = rsqrtf(var + 1e-5f);
    }
}

// ---------------------------------------------------------------------------
// Kernel 4: normalize in place
// ---------------------------------------------------------------------------
__global__ __launch_bounds__(256) void bn_apply(
    float* __restrict__ out, const float* __restrict__ stats,
    const float* __restrict__ gamma, const float* __restrict__ beta)
{
    const int idx = blockIdx.x * 256 + threadIdx.x;
    if (idx < 80000) {
        const float mean = stats[0];
        const float rstd = stats[1];
        out[idx] = (out[idx] - mean) * rstd * gamma[0] + beta[0];
    }
}

// ---------------------------------------------------------------------------
extern "C" void kernel_launch(void* const* d_in, const int* in_sizes, int n_in,
                              void* d_out, int out_size, void* d_ws, size_t ws_size,
                              hipStream_t stream) {
    const float* z     = (const float*)d_in[0];  // [128,512,8,8]
    const float* x     = (const float*)d_in[1];  // [128,512,32,32]
    const float* gamma = (const float*)d_in[2];  // [1]
    const float* beta  = (const float*)d_in[3];  // [1]
    float* out = (float*)d_out;                  // [128,1,25,25] = 80000

    float* P        = (float*)d_ws;              // 128*64*1024 floats = 33.5 MB
    float* blkSum   = P + (size_t)128 * 64 * 1024;
    float* blkSumSq = blkSum + 128;
    float* stats    = blkSumSq + 128;

    xcorr_gemm<<<1024, 256, 0, stream>>>(z, x, P);
    gather_bn <<<128,  256, 0, stream>>>(P, out, blkSum, blkSumSq);
    bn_stats  <<<1,    128, 0, stream>>>(blkSum, blkSumSq, stats);
    bn_apply  <<<(80000 + 255) / 256, 256, 0, stream>>>(out, stats, gamma, beta);
}